// RotateEmbedding_11776800325964
// MI455X (gfx1250) — compile-verified
//
#include <hip/hip_runtime.h>

// RotateEmbedding fused kernel for MI455X (gfx1250, wave32).
// out[t, :] = weight[ids[t], :] @ rot   -- gather-then-rotate (memory bound).
//
// Tiling: block = 256 threads = 8 waves. Block tile = 16 tokens x 128 cols.
// Wave tile = 16x16 via V_WMMA_F32_16X16X4_F32, K staged through LDS in
// chunks of 64.

#define VOCAB 128000
#define DIM   2048
#define TOKENS (4 * 4096)

#define KC    64            // K chunk staged in LDS
#define NTILE 128           // columns per block
#define A_STRIDE (KC + 4)   // 68 floats: 16 A-lanes hit distinct banks
#define B_STRIDE (NTILE + 16) // 144 floats: lo/hi half-wave rows hit disjoint banks

typedef float v2f __attribute__((ext_vector_type(2)));
typedef float v8f __attribute__((ext_vector_type(8)));

__global__ __launch_bounds__(256)
void rotate_embed_wmma(const int* __restrict__ ids,
                       const float* __restrict__ W,
                       const float* __restrict__ R,
                       float* __restrict__ out)
{
    __shared__ float As[16 * A_STRIDE];   //  4.25 KB
    __shared__ float Bs[KC * B_STRIDE];   // 36.0  KB

    const int tid  = threadIdx.x;
    const int wave = tid >> 5;            // 0..7 -> 16-col slice
    const int lane = tid & 31;
    const int hi   = lane >> 4;           // half-wave select (K+2/K+3, M+8)
    const int mn   = lane & 15;           // M for A, N for B/D

    const int n0 = blockIdx.x * NTILE;    // column base
    const int t0 = blockIdx.y * 16;       // token base

    // A staging assignment: thread (tid>>4) owns gathered row, (tid&15) owns
    // one float4 of the 64-float K chunk.
    const int  arow     = tid >> 4;                       // 0..15
    const int  avec     = tid & 15;                       // 0..15
    const long arowbase = (long)ids[t0 + arow] * DIM;     // gathered row

    v8f acc = {};

    for (int k0 = 0; k0 < DIM; k0 += KC) {
        // ---- load staging data to registers (global, coalesced) ----
        const float4 aval =
            *(const float4*)(W + arowbase + k0 + avec * 4);

        float4 bval[8];
        #pragma unroll
        for (int i = 0; i < 8; ++i) {
            const int idx = tid + i * 256;     // 0..2047 float4s of B chunk
            const int kr  = idx >> 5;          // 0..63
            const int nv  = idx & 31;          // 0..31
            bval[i] = *(const float4*)(R + (long)(k0 + kr) * DIM + n0 + nv * 4);
        }

        // prefetch next rot chunk into cache (gfx1250 global_prefetch_b8)
        if (k0 + KC < DIM)
            __builtin_prefetch(R + (long)(k0 + KC) * DIM + n0 + tid, 0, 0);

        __syncthreads();   // previous iteration's LDS reads done

        // ---- write LDS tiles ----
        *(float4*)(&As[arow * A_STRIDE + avec * 4]) = aval;
        #pragma unroll
        for (int i = 0; i < 8; ++i) {
            const int idx = tid + i * 256;
            const int kr  = idx >> 5;
            const int nv  = idx & 31;
            *(float4*)(&Bs[kr * B_STRIDE + nv * 4]) = bval[i];
        }

        __syncthreads();   // tiles visible to all waves

        // ---- 16 WMMA steps of K=4 each ----
        #pragma unroll
        for (int kk = 0; kk < KC; kk += 4) {
            // A operand (16x4 f32): lane holds M=mn; lo half K=kk,kk+1,
            // hi half K=kk+2,kk+3 -> contiguous pair, one ds_load_b64.
            v2f a = *(const v2f*)(&As[mn * A_STRIDE + kk + 2 * hi]);

            // B operand (4x16 f32): row K striped across lanes, N=mn;
            // lo half rows kk/kk+1, hi half rows kk+2/kk+3.
            v2f b;
            b.x = Bs[(kk + 2 * hi + 0) * B_STRIDE + wave * 16 + mn];
            b.y = Bs[(kk + 2 * hi + 1) * B_STRIDE + wave * 16 + mn];

            acc = __builtin_amdgcn_wmma_f32_16x16x4_f32(
                /*neg_a=*/false, a, /*neg_b=*/false, b,
                /*c_mod=*/(short)0, acc,
                /*reuse_a=*/false, /*reuse_b=*/false);
        }
    }

    // ---- store D: VGPR r -> row M = r + 8*hi, col N = mn (ISA 7.12.2) ----
    const int col = n0 + wave * 16 + mn;
    #pragma unroll
    for (int r = 0; r < 8; ++r) {
        out[(long)(t0 + r + 8 * hi) * DIM + col] = acc[r];
    }
}

extern "C" void kernel_launch(void* const* d_in, const int* in_sizes, int n_in,
                              void* d_out, int out_size, void* d_ws, size_t ws_size,
                              hipStream_t stream) {
    (void)in_sizes; (void)n_in; (void)out_size; (void)d_ws; (void)ws_size;

    const int*   ids    = (const int*)  d_in[0];  // [4, 4096] int32 token ids
    const float* weight = (const float*)d_in[1];  // [128000, 2048] f32
    const float* rot    = (const float*)d_in[2];  // [2048, 2048] f32
    float*       out    = (float*)      d_out;    // [4, 4096, 2048] f32

    dim3 grid(DIM / NTILE, TOKENS / 16);  // (16, 1024)
    rotate_embed_wmma<<<grid, 256, 0, stream>>>(ids, weight, rot, out);
}